// kernel_generated_1_21887153341172
// MI455X (gfx1250) — compile-verified
//
#include <hip/hip_runtime.h>

typedef __attribute__((ext_vector_type(2))) float v2f;
typedef __attribute__((ext_vector_type(8))) float v8f;

// Stage 1: A[128,784] = W[128,192] x X[192,784] via V_WMMA_F32_16X16X4_F32.
// K padded 192 -> 256 (q' = j*4 + k, slot k==3 zero) so each K-step of 4 is
// exactly one input channel j == kt: per-lane K phase (k = 2*half + g) is
// loop-invariant, addresses advance by constant strides, and padding is a
// precomputed 0/1 multiply (all loads in-bounds via clamped indices).
// The k==3 slot only needs zeroing on the A side: D += A x B means the slot
// contributes A[m,3]*B[3,n], and A[.,3] == 0 suffices, so B's g=1 element
// needs no mask at all.
__global__ __launch_bounds__(32) void conv_gemm_wmma(
    const float* __restrict__ x,    // (64, 28, 28)
    const float* __restrict__ w1,   // (128, 64, 3) == (128, 192)
    float* __restrict__ A)          // (128, 784) workspace
{
    const int tile = blockIdx.x;          // 0..391  (8 * 49 tiles)
    const int mt   = tile & 7;            // i-tile (8 x 16 over M=128)
    const int nt   = tile >> 3;           // p-tile (49 x 16 over N=784)
    const int lane = threadIdx.x & 31;
    const int mn   = lane & 15;           // row (A frag) / col (B, C/D frags)
    const int half = lane >> 4;
    const int c2   = half * 2;            // per-lane K phase base: 0 or 2

    // A-fragment source row of W (w1 viewed [128,192], q = j*3 + k)
    const float* __restrict__ wrow = w1 + (mt * 16 + mn) * 192;

    // B/D column (flattened output position p = h*28 + w)
    const int p    = nt * 16 + mn;
    const int h    = p / 28;
    const int w    = p - h * 28;
    const int xrow = h * 28;

    // ---- loop-invariant per-lane offsets & padding masks ----
    // A g=0: k = c2 in {0,2}     -> wrow[3*kt + c2], always valid, no mask.
    // A g=1: k = c2+1 in {1,3};  k==3 is the zero pad slot (hi_m).
    const int   ax_off = c2;
    const int   ay_off = (c2 + 1 > 2) ? 2 : (c2 + 1);   // clamped safe offset
    const float hi_m   = (half == 0) ? 1.0f : 0.0f;     // zeroes A's k==3 slot

    // B g=0: k = c2 taps x[.., w + c2 - 1] (w-1 or w+1): boundary mask + clamp.
    int we0 = w + c2 - 1;
    const float bx_m = ((unsigned)we0 < 28u) ? 1.0f : 0.0f;
    we0 = we0 < 0 ? 0 : (we0 > 27 ? 27 : we0);
    const int bx_off = xrow + we0;
    // B g=1: half=0 -> k=1 taps x[.., w] (always valid); half=1 -> k=3 slot,
    // value irrelevant because A[.,3] == 0. No mask needed.
    const int by_off = xrow + w;

    v8f acc = {};
    #pragma unroll 16
    for (int kt = 0; kt < 64; ++kt) {     // kt == input channel j
        const float* __restrict__ xk = x + kt * 784;
        const float a0 = wrow[3 * kt + ax_off];
        const float a1 = wrow[3 * kt + ay_off] * hi_m;
        const float b0 = xk[bx_off] * bx_m;
        const float b1 = xk[by_off];
        v2f av = {a0, a1};
        v2f bv = {b0, b1};
        acc = __builtin_amdgcn_wmma_f32_16x16x4_f32(
                /*neg_a=*/false, av, /*neg_b=*/false, bv,
                /*c_mod=*/(short)0, acc, /*reuse_a=*/false, /*reuse_b=*/false);
    }

    // Store D: VGPR r holds D[r + 8*half, mn]  (ISA 16x16 f32 C/D layout)
    #pragma unroll
    for (int r = 0; r < 8; ++r) {
        const int iout = mt * 16 + r + 8 * half;
        A[iout * 784 + p] = acc[r];
    }
}

// Stage 2: out[i,h,w] = w0[i,0]*A[i,(h-1)%28,w] + w0[i,1]*A[i,(h-2)%28,w]
// (the roll/reshape/roll/sum of the reference collapses to this).
__global__ __launch_bounds__(256) void roll_combine(
    const float* __restrict__ A,    // (128, 784)
    const float* __restrict__ w0,   // (128, 2)
    float* __restrict__ out)        // (128, 28, 28)
{
    const int tid = blockIdx.x * 256 + threadIdx.x;
    if (tid >= 128 * 784) return;
    const int i = tid / 784;
    const int p = tid - i * 784;
    const int h = p / 28;
    const int w = p - h * 28;
    int h1 = h - 1; if (h1 < 0) h1 += 28;
    int h2 = h - 2; if (h2 < 0) h2 += 28;
    out[tid] = w0[2 * i + 0] * A[i * 784 + h1 * 28 + w]
             + w0[2 * i + 1] * A[i * 784 + h2 * 28 + w];
}

extern "C" void kernel_launch(void* const* d_in, const int* in_sizes, int n_in,
                              void* d_out, int out_size, void* d_ws, size_t ws_size,
                              hipStream_t stream) {
    (void)in_sizes; (void)n_in; (void)out_size; (void)ws_size;
    const float* x  = (const float*)d_in[0];   // (1,64,28,28)
    const float* w0 = (const float*)d_in[1];   // (128,2)
    const float* w1 = (const float*)d_in[2];   // (128,64,3)
    float* out = (float*)d_out;                // (1,128,28,28)
    float* A   = (float*)d_ws;                 // needs 128*784*4 = 401,408 B

    conv_gemm_wmma<<<392, 32, 0, stream>>>(x, w1, A);
    roll_combine<<<(128 * 784 + 255) / 256, 256, 0, stream>>>(A, w0, out);
}